// SigNet2_11038065951552
// MI455X (gfx1250) — compile-verified
//
#include <hip/hip_runtime.h>

typedef float v2f __attribute__((ext_vector_type(2)));
typedef float v8f __attribute__((ext_vector_type(8)));

#define B_BATCH 64
#define S_IN    256
#define C_IN    13
#define N_STEPS 253      // S - (KERNEL-1)
#define C       16       // augmented channels
#define SIG123N 4368     // 16 + 256 + 4096
#define SIG4N   65536
#define SIGC    69904
#define OUTN    512

// ---------------------------------------------------------------------------
// Kernel 1: augment + increments.  One block per batch, thread t = time step.
// dv[b,t,:] = x[b,t,:] - x[b,t-1,:]  (dv[b,0,:] = x[b,0,:])
// ---------------------------------------------------------------------------
__device__ inline void compute_x_row(const float* __restrict__ inp_b, int t,
                                     const float* sw1, const float* sb1,
                                     const float* sw2, const float* sb2,
                                     const float* sw3, const float* sb3,
                                     float* x) {
  float h1[8];
#pragma unroll
  for (int o = 0; o < 8; ++o) {
    float a = sb1[o];
#pragma unroll
    for (int i = 0; i < C_IN; ++i)
#pragma unroll
      for (int k = 0; k < 4; ++k)
        a += sw1[(o * C_IN + i) * 4 + k] * inp_b[(t + k) * C_IN + i];
    h1[o] = a > 0.f ? a : 0.f;               // relu before second conv
  }
  float h2[8];
#pragma unroll
  for (int o = 0; o < 8; ++o) {
    float a = sb2[o];
#pragma unroll
    for (int j = 0; j < 8; ++j) a += sw2[o * 8 + j] * h1[j];
    h2[o] = a > 0.f ? a : 0.f;               // relu before third conv
  }
  float h3[2];
#pragma unroll
  for (int o = 0; o < 2; ++o) {
    float a = sb3[o];
#pragma unroll
    for (int j = 0; j < 8; ++j) a += sw3[o * 8 + j] * h2[j];
    h3[o] = a;
  }
#pragma unroll
  for (int c = 0; c < C_IN; ++c) x[c] = inp_b[(t + 3) * C_IN + c];
  x[13] = (float)t * (1.0f / 252.0f);
  x[14] = h3[0];
  x[15] = h3[1];
}

__global__ __launch_bounds__(256)
void augment_kernel(const float* __restrict__ inp,
                    const float* __restrict__ w1, const float* __restrict__ b1,
                    const float* __restrict__ w2, const float* __restrict__ b2,
                    const float* __restrict__ w3, const float* __restrict__ b3,
                    float* __restrict__ DV) {
  __shared__ float sw1[8 * C_IN * 4], sb1[8], sw2[64], sb2[8], sw3[16], sb3[2];
  const int tid = threadIdx.x;
  for (int i = tid; i < 8 * C_IN * 4; i += 256) sw1[i] = w1[i];
  if (tid < 64) sw2[tid] = w2[tid];
  if (tid < 16) sw3[tid] = w3[tid];
  if (tid < 8)  { sb1[tid] = b1[tid]; sb2[tid] = b2[tid]; }
  if (tid < 2)  sb3[tid] = b3[tid];
  __syncthreads();

  const int b = blockIdx.x;
  const int t = tid;
  if (t < N_STEPS) {
    const float* inp_b = inp + (size_t)b * S_IN * C_IN;
    float xt[C];
    compute_x_row(inp_b, t, sw1, sb1, sw2, sb2, sw3, sb3, xt);
    float dv[C];
    if (t == 0) {
#pragma unroll
      for (int c = 0; c < C; ++c) dv[c] = xt[c];
    } else {
      float xp[C];
      compute_x_row(inp_b, t - 1, sw1, sb1, sw2, sb2, sw3, sb3, xp);
#pragma unroll
      for (int c = 0; c < C; ++c) dv[c] = xt[c] - xp[c];
    }
    float* out = DV + ((size_t)b * N_STEPS + t) * C;
#pragma unroll
    for (int c = 0; c < C; ++c) out[c] = dv[c];
  }
}

// ---------------------------------------------------------------------------
// Kernel 2: sequential signature scan.  One 1024-thread block per batch.
// Horner form:  t4 = sig3 + (sig2 + (sig1 + dv/4) (x) dv/3) (x) dv/2
//               sig4 += T4_panel(4096x4) @ DV_panel(4x16)  via WMMA f32 16x16x4
// sig4 lives entirely in WMMA accumulators (32 waves x 8 tiles).
// ---------------------------------------------------------------------------
__global__ __launch_bounds__(1024)
void sig_scan_kernel(const float* __restrict__ DV,
                     float* __restrict__ SIG123,
                     float* __restrict__ SIG4) {
  extern __shared__ float sm[];
  float* sDV   = sm;             // 4 x 16 current dv panel
  float* sSig1 = sm + 64;        // 16
  float* sSig2 = sm + 80;        // 256
  float* sS    = sm + 336;       // 256  (mid tensor for t4)
  float* sT3   = sm + 592;       // 256
  float* sSig3 = sm + 848;       // 4096
  float* sT4   = sm + 4944;      // 4096 x 4, row-major stride 4

  const int tid  = threadIdx.x;
  const int b    = blockIdx.x;
  const int lane = tid & 31;
  const int wave = tid >> 5;     // 0..31
  const int g    = lane >> 4;    // lane half
  const int ln   = lane & 15;

  if (tid < 16)  sSig1[tid] = 0.f;
  if (tid < 256) sSig2[tid] = 0.f;
  for (int e = tid; e < 4096; e += 1024) sSig3[e] = 0.f;

  v8f acc[8];
#pragma unroll
  for (int t = 0; t < 8; ++t) acc[t] = (v8f){};
  __syncthreads();

  const float* dvb = DV + (size_t)b * N_STEPS * C;

  for (int k = 0; k < 256; ++k) {         // 64 chunks of 4; steps >= 253 padded
    const int kk = k & 3;
    if (tid < 16) sDV[kk * 16 + tid] = (k < N_STEPS) ? dvb[k * 16 + tid] : 0.f;
    __syncthreads();

    if (tid < 256) {                      // level-2-sized tensors
      const int i = tid >> 4, j = tid & 15;
      const float dvi = sDV[kk * 16 + i];
      const float dvj = sDV[kk * 16 + j];
      const float s1  = sSig1[i];
      const float s2  = sSig2[tid];
      sS[tid]    = s2 + (s1 + dvi * 0.25f) * dvj * (1.0f / 3.0f);
      sT3[tid]   = s2 + (s1 + dvi * (1.0f / 3.0f)) * dvj * 0.5f;
      sSig2[tid] = s2 + (s1 + dvi * 0.5f) * dvj;    // level-2 update (old sig1)
    }
    __syncthreads();

#pragma unroll
    for (int p = 0; p < 4; ++p) {         // level-3-sized tensors: t4 + sig3
      const int e  = tid + p * 1024;
      const int ij = e >> 4, k3 = e & 15;
      const float dvk = sDV[kk * 16 + k3];
      const float s3  = sSig3[e];
      sT4[e * 4 + kk] = s3 + sS[ij] * dvk * 0.5f;   // column kk of T4 panel
      sSig3[e]        = s3 + sT3[ij] * dvk;         // level-3 update
    }
    if (tid < 16) sSig1[tid] += sDV[kk * 16 + tid]; // level-1 update
    __syncthreads();

    if (kk == 3) {                        // flush panel: sig4 += T4 @ DVp
      v2f bf;                             // B 4x16: K = v + 2g, N = ln
      bf.x = sDV[(0 + 2 * g) * 16 + ln];
      bf.y = sDV[(1 + 2 * g) * 16 + ln];
#pragma unroll
      for (int t = 0; t < 8; ++t) {       // A 16x4: M = ln, K = v + 2g
        const int row = (wave * 8 + t) * 16 + ln;
        v2f af;
        af.x = sT4[row * 4 + 0 + 2 * g];
        af.y = sT4[row * 4 + 1 + 2 * g];
        acc[t] = __builtin_amdgcn_wmma_f32_16x16x4_f32(
            false, af, false, bf, (short)0, acc[t], false, false);
      }
      __syncthreads();                    // panel consumed before next writes
    }
  }

  // write out sig1|sig2|sig3 (contiguous, matches W column order)
  float* s123 = SIG123 + (size_t)b * SIG123N;
  if (tid < 16)  s123[tid] = sSig1[tid];
  if (tid < 256) s123[16 + tid] = sSig2[tid];
  for (int e = tid; e < 4096; e += 1024) s123[272 + e] = sSig3[e];

  // write sig4 from accumulators.  C layout: M = v + 8g, N = ln
  float* s4 = SIG4 + (size_t)b * SIG4N;
#pragma unroll
  for (int t = 0; t < 8; ++t) {
    const int mt = wave * 8 + t;
#pragma unroll
    for (int v = 0; v < 8; ++v) {
      const int m = mt * 16 + v + 8 * g;
      s4[m * 16 + ln] = acc[t][v];
    }
  }
}

// ---------------------------------------------------------------------------
// Kernel 3: out(64x512) = [sig123|sig4](64x69904) @ W^T + bL.   f32 WMMA.
// 128 waves, one 16x16 output tile each.  W is L2-resident (143MB < 192MB).
// ---------------------------------------------------------------------------
__global__ __launch_bounds__(256)
void final_gemm_kernel(const float* __restrict__ SIG123,
                       const float* __restrict__ SIG4,
                       const float* __restrict__ W,
                       const float* __restrict__ bL,
                       float* __restrict__ out) {
  const int tid   = threadIdx.x;
  const int lane  = tid & 31;
  const int wave  = tid >> 5;
  const int gw    = blockIdx.x * 8 + wave;   // 0..127
  const int mtile = gw >> 5;                 // 0..3   (batch / 16)
  const int ntile = gw & 31;                 // 0..31  (out / 16)
  const int g     = lane >> 4;
  const int ln    = lane & 15;

  const int mrow = mtile * 16 + ln;          // batch row this lane loads (A)
  const int orow = ntile * 16 + ln;          // W row this lane loads (B)
  const float* Wr = W + (size_t)orow * SIGC;

  v8f acc = (v8f){};

  // segment 1: sig1|sig2|sig3, K = 4368
  const float* A1 = SIG123 + (size_t)mrow * SIG123N;
#pragma unroll 4
  for (int k0 = 0; k0 < SIG123N; k0 += 4) {
    v2f a, bb;
    a.x  = A1[k0 + 0 + 2 * g];
    a.y  = A1[k0 + 1 + 2 * g];
    bb.x = Wr[k0 + 0 + 2 * g];
    bb.y = Wr[k0 + 1 + 2 * g];
    acc = __builtin_amdgcn_wmma_f32_16x16x4_f32(
        false, a, false, bb, (short)0, acc, false, false);
  }

  // segment 2: sig4, K = 65536, W columns offset by 4368
  const float* A2 = SIG4 + (size_t)mrow * SIG4N;
  const float* W2 = Wr + SIG123N;
#pragma unroll 4
  for (int k0 = 0; k0 < SIG4N; k0 += 4) {
    v2f a, bb;
    a.x  = A2[k0 + 0 + 2 * g];
    a.y  = A2[k0 + 1 + 2 * g];
    bb.x = W2[k0 + 0 + 2 * g];
    bb.y = W2[k0 + 1 + 2 * g];
    acc = __builtin_amdgcn_wmma_f32_16x16x4_f32(
        false, a, false, bb, (short)0, acc, false, false);
  }

  const int o = ntile * 16 + ln;
  const float bias = bL[o];
#pragma unroll
  for (int v = 0; v < 8; ++v) {
    const int brow = mtile * 16 + v + 8 * g;  // C layout: M = v + 8g
    out[(size_t)brow * OUTN + o] = acc[v] + bias;
  }
}

// ---------------------------------------------------------------------------
extern "C" void kernel_launch(void* const* d_in, const int* in_sizes, int n_in,
                              void* d_out, int out_size, void* d_ws, size_t ws_size,
                              hipStream_t stream) {
  const float* inp = (const float*)d_in[0];
  const float* w1  = (const float*)d_in[1];
  const float* b1  = (const float*)d_in[2];
  const float* w2  = (const float*)d_in[3];
  const float* b2  = (const float*)d_in[4];
  const float* w3  = (const float*)d_in[5];
  const float* b3  = (const float*)d_in[6];
  const float* W   = (const float*)d_in[7];
  const float* bL  = (const float*)d_in[8];
  float* out = (float*)d_out;

  float* ws     = (float*)d_ws;
  float* DV     = ws;                                   // 64*253*16  = 259072 f
  float* SIG123 = ws + (size_t)B_BATCH * N_STEPS * C;   // 64*4368    = 279552 f
  float* SIG4   = SIG123 + (size_t)B_BATCH * SIG123N;   // 64*65536   = 4194304 f

  const int smem_scan = (64 + 16 + 256 + 256 + 256 + 4096 + 16384) * 4; // 85312 B
  (void)hipFuncSetAttribute((const void*)sig_scan_kernel,
                            hipFuncAttributeMaxDynamicSharedMemorySize,
                            smem_scan);

  augment_kernel<<<B_BATCH, 256, 0, stream>>>(inp, w1, b1, w2, b2, w3, b3, DV);
  sig_scan_kernel<<<B_BATCH, 1024, smem_scan, stream>>>(DV, SIG123, SIG4);
  final_gemm_kernel<<<16, 256, 0, stream>>>(SIG123, SIG4, W, bL, out);
}